// GatedLinearAttention_65755949302270
// MI455X (gfx1250) — compile-verified
//
#include <hip/hip_runtime.h>

// ---------------------------------------------------------------------------
// Gated linear attention, MI455X (gfx1250), all-FP32 WMMA path.
// B=4, N=4096, C=1024, H=16, D=64. HBM-bound (~0.5GB traffic @ 23.3TB/s),
// so fp32 V_WMMA_F32_16X16X4_F32 is already far past the bandwidth roofline.
// v3: operand staging now uses the CDNA5 direct global->LDS async path
// (GLOBAL_LOAD_ASYNC_TO_LDS_B128, ASYNCcnt) instead of load->VGPR->ds_store;
// inner loops remain ds_load + v_wmma only.
// ---------------------------------------------------------------------------

typedef __attribute__((ext_vector_type(2))) float v2f;
typedef __attribute__((ext_vector_type(8))) float v8f;

constexpr int Bc = 4;
constexpr int Nc = 4096;
constexpr int Cc = 1024;
constexpr int Hc = 16;
constexpr int Dc = 64;
constexpr int K3C = 3 * Cc;

__device__ __forceinline__ v8f wmma_f32(v2f a, v2f b, v8f c) {
    // D = A(16x4) * B(4x16) + C(16x16), all fp32
    return __builtin_amdgcn_wmma_f32_16x16x4_f32(
        /*neg_a=*/false, a, /*neg_b=*/false, b,
        /*c_mod=*/(short)0, c, /*reuse_a=*/false, /*reuse_b=*/false);
}

// elu(x)+1 : x>0 ? x+1 : exp(x)
__device__ __forceinline__ float elu1(float x) {
    return x > 0.0f ? (x + 1.0f) : __expf(x);
}

// Direct global -> LDS async copy of 16 bytes (CDNA5 TDM-adjacent path,
// tracked by ASYNCcnt). vdst operand is the LDS byte offset (low 32 bits of
// the flat shared-aperture address per ISA 10.2); vaddr is a 64-bit VGPR pair.
__device__ __forceinline__ void async_ld16(void* lds_dst, const void* gsrc) {
    const unsigned ldsoff = (unsigned)(uintptr_t)lds_dst;
    asm volatile("global_load_async_to_lds_b128 %0, %1, off"
                 :: "v"(ldsoff), "v"(gsrc)
                 : "memory");
}

__device__ __forceinline__ void wait_async() {
    asm volatile("s_wait_asynccnt 0x0" ::: "memory");
}

__global__ void zero_f32_kernel(float* __restrict__ p, int n) {
    int i = blockIdx.x * 256 + threadIdx.x;
    if (i < n) p[i] = 0.0f;
}

// ---------------------------------------------------------------------------
// Kernel 1: fused qkv + gate projection.
// Block = 128 threads = 4 waves. Wave 0->q, 1->k, 2->v, 3->gate; all four
// share the x tile in LDS; each wave async-stages its own 32x16 weight tile.
// Epilogue: gate->LDS, sigmoid gating + elu+1 on q,k; k_sum via atomics.
// grid = (C/16, (B*N)/16)
// ---------------------------------------------------------------------------
__global__ __launch_bounds__(128) void proj_gate_kernel(
    const float* __restrict__ x, const float* __restrict__ Wqkv,
    const float* __restrict__ Wg, const float* __restrict__ bg,
    float* __restrict__ qb, float* __restrict__ kb, float* __restrict__ vb,
    float* __restrict__ ksum)
{
    __shared__ float As[16][36];       // 16 rows x 32 K (+pad, 16B-aligned rows)
    __shared__ float Bs[4][32][20];    // per-wave 32 K x 16 cols (+pad)
    __shared__ float Gs[16][17];       // sigmoid gate tile

    const int tid   = threadIdx.x;
    const int wave  = tid >> 5;
    const int lane  = tid & 31;
    const int l15   = lane & 15;
    const int khalf = (lane >> 4) * 2;   // K-pair select per ISA A-layout
    const int row0  = blockIdx.y * 16;
    const int col0  = blockIdx.x * 16;

    const float* W;
    int ldw, wcol0;
    if (wave == 0)      { W = Wqkv; ldw = K3C; wcol0 = col0; }
    else if (wave == 1) { W = Wqkv; ldw = K3C; wcol0 = Cc + col0; }
    else if (wave == 2) { W = Wqkv; ldw = K3C; wcol0 = 2 * Cc + col0; }
    else                { W = Wg;   ldw = Cc;  wcol0 = col0; }

    v8f acc = {};
    for (int k0 = 0; k0 < Cc; k0 += 32) {
        // x tile: 16x32 floats, one async b128 per thread
        {
            const int r  = tid >> 3;
            const int cg = (tid & 7) * 4;
            const float* src = x + (size_t)(row0 + r) * Cc + k0 + cg;
            __builtin_prefetch(src + 32, 0, 1);
            async_ld16(&As[r][cg], src);
        }
        // per-wave weight tile: lane owns K-row `lane`, 16 cols = 4x async b128
        {
            const float* wsrc = W + (size_t)(k0 + lane) * ldw + wcol0;
            #pragma unroll
            for (int t = 0; t < 4; ++t)
                async_ld16(&Bs[wave][lane][t * 4], wsrc + t * 4);
        }
        wait_async();
        __syncthreads();
        #pragma unroll
        for (int kk = 0; kk < 32; kk += 4) {
            v2f a, b;
            a.x = As[l15][kk + khalf + 0];
            a.y = As[l15][kk + khalf + 1];
            b.x = Bs[wave][kk + khalf + 0][l15];
            b.y = Bs[wave][kk + khalf + 1][l15];
            acc = wmma_f32(a, b, acc);
        }
        __syncthreads();
    }

    // wave 3 publishes sigmoid(gate + bg) to LDS
    if (wave == 3) {
        const float bgi = bg[col0 + l15];
        #pragma unroll
        for (int r = 0; r < 8; ++r) {
            const int m = r + (lane >> 4) * 8;
            Gs[m][l15] = 1.0f / (1.0f + __expf(-(acc[r] + bgi)));
        }
    }
    __syncthreads();

    if (wave == 0) {                       // q' = elu(q * gate) + 1
        #pragma unroll
        for (int r = 0; r < 8; ++r) {
            const int m = r + (lane >> 4) * 8;
            qb[(size_t)(row0 + m) * Cc + col0 + l15] = elu1(acc[r] * Gs[m][l15]);
        }
    } else if (wave == 1) {                // k' = elu(k * gate) + 1, + k_sum
        float psum = 0.0f;
        #pragma unroll
        for (int r = 0; r < 8; ++r) {
            const int m = r + (lane >> 4) * 8;
            const float kv = elu1(acc[r] * Gs[m][l15]);
            kb[(size_t)(row0 + m) * Cc + col0 + l15] = kv;
            psum += kv;
        }
        const int b = row0 / Nc;           // 16-row tile never crosses batch
        atomicAdd(&ksum[b * Cc + col0 + l15], psum);
    } else if (wave == 2) {                // v passthrough
        #pragma unroll
        for (int r = 0; r < 8; ++r) {
            const int m = r + (lane >> 4) * 8;
            vb[(size_t)(row0 + m) * Cc + col0 + l15] = acc[r];
        }
    }
}

// ---------------------------------------------------------------------------
// Kernel 2: kv[b,h] = k'^T @ v  (64x64 over N=4096) per head.
// Block = 4 waves; wave w owns e-tile w. Both k' (64x16) and v (64x64)
// chunks async-staged in LDS; inner 16-WMMA loop is LDS-only.
// grid = (4 d-tiles, B*H)
// ---------------------------------------------------------------------------
__global__ __launch_bounds__(128) void kv_state_kernel(
    const float* __restrict__ kb, const float* __restrict__ vb,
    float* __restrict__ kvb)
{
    __shared__ float Ks[64][20];   // 64 n x 16 d (+pad)
    __shared__ float Vs[64][68];   // 64 n x 64 e (+pad)

    const int tid   = threadIdx.x;
    const int wave  = tid >> 5;
    const int lane  = tid & 31;
    const int l15   = lane & 15;
    const int khalf = (lane >> 4) * 2;
    const int d0    = blockIdx.x * 16;
    const int bh    = blockIdx.y;
    const int b     = bh >> 4, h = bh & 15;
    const int e0    = wave * 16;
    const size_t base = (size_t)b * Nc * Cc + h * 64;

    v8f acc = {};
    for (int n0 = 0; n0 < Nc; n0 += 64) {
        {   // k' chunk: 64 x 16, 2x async b128 per lane
            const int nl = tid >> 1;
            const int cg = (tid & 1) * 8;
            const float* src = kb + base + (size_t)(n0 + nl) * Cc + d0 + cg;
            async_ld16(&Ks[nl][cg + 0], src + 0);
            async_ld16(&Ks[nl][cg + 4], src + 4);
        }
        {   // v chunk: 64 x 64, 8x async b128 per lane
            #pragma unroll
            for (int t = 0; t < 8; ++t) {
                const int idx = tid * 8 + t;        // float4 index 0..1023
                const int rr  = idx >> 4;           // 16 float4 per row
                const int cg  = (idx & 15) * 4;
                const float* src = vb + base + (size_t)(n0 + rr) * Cc + cg;
                async_ld16(&Vs[rr][cg], src);
            }
        }
        wait_async();
        __syncthreads();
        #pragma unroll 4
        for (int kk = 0; kk < 64; kk += 4) {
            v2f a, bv;
            a.x  = Ks[kk + khalf + 0][l15];         // A[d][n] = k'[n][d]
            a.y  = Ks[kk + khalf + 1][l15];
            bv.x = Vs[kk + khalf + 0][e0 + l15];
            bv.y = Vs[kk + khalf + 1][e0 + l15];
            acc = wmma_f32(a, bv, acc);
        }
        __syncthreads();
    }
    #pragma unroll
    for (int r = 0; r < 8; ++r) {
        const int m = r + (lane >> 4) * 8;
        kvb[((size_t)bh * 64 + d0 + m) * 64 + e0 + l15] = acc[r];
    }
}

// ---------------------------------------------------------------------------
// Kernel 3: out = (q' @ kv) / (q'.k_sum + 1e-8) per head.
// kv (64x64), k_sum, and q tile (16x64) async-cached in LDS; normalizer per
// 16-row tile computed from LDS. grid = (N/16, B*H)
// ---------------------------------------------------------------------------
__global__ __launch_bounds__(128) void attn_out_kernel(
    const float* __restrict__ qb, const float* __restrict__ kvb,
    const float* __restrict__ ksum, float* __restrict__ attn)
{
    __shared__ float KVs[64][64];
    __shared__ float Qs[16][68];
    __shared__ float Sum[64];
    __shared__ float Nrm[16];

    const int tid   = threadIdx.x;
    const int wave  = tid >> 5;
    const int lane  = tid & 31;
    const int l15   = lane & 15;
    const int khalf = (lane >> 4) * 2;
    const int n0    = blockIdx.x * 16;
    const int bh    = blockIdx.y;
    const int b     = bh >> 4, h = bh & 15;
    const int e0    = wave * 16;
    const size_t qbase = (size_t)b * Nc * Cc + h * 64;

    {   // cache kv state, q tile (async b128), k_sum
        const float* kvsrc = kvb + (size_t)bh * 64 * 64;
        #pragma unroll
        for (int t = 0; t < 8; ++t) {
            const int idx = tid * 8 + t;            // float4 index 0..1023
            async_ld16(&((float*)KVs)[idx * 4], kvsrc + idx * 4);
        }
        #pragma unroll
        for (int t = 0; t < 2; ++t) {
            const int idx = tid * 2 + t;            // float4 index 0..255
            const int rr  = idx >> 4;               // 16 float4 per row
            const int cg  = (idx & 15) * 4;
            const float* src = qb + qbase + (size_t)(n0 + rr) * Cc + cg;
            async_ld16(&Qs[rr][cg], src);
        }
        if (tid < 64) Sum[tid] = ksum[b * Cc + h * 64 + tid];
    }
    wait_async();
    __syncthreads();

    if (tid < 16) {   // normalizer for the 16 rows of this tile (from LDS)
        float s = 0.0f;
        #pragma unroll 8
        for (int d = 0; d < 64; ++d) s += Qs[tid][d] * Sum[d];
        Nrm[tid] = s + 1e-8f;
    }
    __syncthreads();

    v8f acc = {};
    #pragma unroll
    for (int kk = 0; kk < 64; kk += 4) {
        v2f a, bv;
        a.x  = Qs[l15][kk + khalf + 0];
        a.y  = Qs[l15][kk + khalf + 1];
        bv.x = KVs[kk + khalf + 0][e0 + l15];
        bv.y = KVs[kk + khalf + 1][e0 + l15];
        acc = wmma_f32(a, bv, acc);
    }
    #pragma unroll
    for (int r = 0; r < 8; ++r) {
        const int m = r + (lane >> 4) * 8;
        attn[qbase + (size_t)(n0 + m) * Cc + e0 + l15] = acc[r] / Nrm[m];
    }
}

// ---------------------------------------------------------------------------
// Kernel 4: final projection out = attn @ Wp + bp.
// Block = 4 waves, 16 rows x 64 cols; A tile and the 32x64 Wp slab both
// async-staged in LDS; each wave reads its 16-col slice.
// grid = (C/64, (B*N)/16)
// ---------------------------------------------------------------------------
__global__ __launch_bounds__(128) void final_proj_kernel(
    const float* __restrict__ attn, const float* __restrict__ Wp,
    const float* __restrict__ bp, float* __restrict__ out)
{
    __shared__ float As[16][36];     // 16 rows x 32 K
    __shared__ float Bs[32][68];     // 32 K x 64 cols (+pad)

    const int tid   = threadIdx.x;
    const int wave  = tid >> 5;
    const int lane  = tid & 31;
    const int l15   = lane & 15;
    const int khalf = (lane >> 4) * 2;
    const int row0  = blockIdx.y * 16;
    const int cbase = blockIdx.x * 64;
    const int colw  = cbase + wave * 16;

    v8f acc = {};
    for (int k0 = 0; k0 < Cc; k0 += 32) {
        {   // A tile: 16 x 32, one async b128 per thread
            const int r  = tid >> 3;
            const int cg = (tid & 7) * 4;
            const float* src = attn + (size_t)(row0 + r) * Cc + k0 + cg;
            __builtin_prefetch(src + 32, 0, 1);
            async_ld16(&As[r][cg], src);
        }
        {   // Wp slab: 32 x 64, 4x async b128 per thread
            #pragma unroll
            for (int t = 0; t < 4; ++t) {
                const int idx = tid * 4 + t;        // float4 index 0..511
                const int rr  = idx >> 4;           // 16 float4 per row
                const int cg  = (idx & 15) * 4;
                const float* src = Wp + (size_t)(k0 + rr) * Cc + cbase + cg;
                async_ld16(&Bs[rr][cg], src);
            }
        }
        wait_async();
        __syncthreads();
        #pragma unroll
        for (int kk = 0; kk < 32; kk += 4) {
            v2f a, b;
            a.x = As[l15][kk + khalf + 0];
            a.y = As[l15][kk + khalf + 1];
            b.x = Bs[kk + khalf + 0][wave * 16 + l15];
            b.y = Bs[kk + khalf + 1][wave * 16 + l15];
            acc = wmma_f32(a, b, acc);
        }
        __syncthreads();
    }
    const float bias = bp[colw + l15];
    #pragma unroll
    for (int r = 0; r < 8; ++r) {
        const int m = r + (lane >> 4) * 8;
        out[(size_t)(row0 + m) * Cc + colw + l15] = acc[r] + bias;
    }
}

// ---------------------------------------------------------------------------
extern "C" void kernel_launch(void* const* d_in, const int* in_sizes, int n_in,
                              void* d_out, int out_size, void* d_ws, size_t ws_size,
                              hipStream_t stream) {
    const float* x    = (const float*)d_in[0];
    const float* Wqkv = (const float*)d_in[1];
    const float* Wg   = (const float*)d_in[2];
    const float* bg   = (const float*)d_in[3];
    const float* Wp   = (const float*)d_in[4];
    const float* bp   = (const float*)d_in[5];
    float* out = (float*)d_out;

    const size_t MC = (size_t)Bc * Nc * Cc;   // 16M floats
    float* ws   = (float*)d_ws;
    float* qb   = ws;                 // q'  (B,N,C)
    float* kb   = qb + MC;            // k'  (B,N,C)
    float* vb   = kb + MC;            // v   (B,N,C)
    float* attn = vb + MC;            // attention output (B,N,C)
    float* kvb  = attn + MC;          // kv state (B*H, 64, 64)
    float* ksum = kvb + (size_t)Bc * Hc * Dc * Dc;   // (B, C)

    zero_f32_kernel<<<(Bc * Cc + 255) / 256, 256, 0, stream>>>(ksum, Bc * Cc);

    proj_gate_kernel<<<dim3(Cc / 16, (Bc * Nc) / 16), 128, 0, stream>>>(
        x, Wqkv, Wg, bg, qb, kb, vb, ksum);

    kv_state_kernel<<<dim3(Dc / 16, Bc * Hc), 128, 0, stream>>>(kb, vb, kvb);

    attn_out_kernel<<<dim3(Nc / 16, Bc * Hc), 128, 0, stream>>>(qb, kvb, ksum, attn);

    final_proj_kernel<<<dim3(Cc / 64, (Bc * Nc) / 16), 128, 0, stream>>>(
        attn, Wp, bp, out);
}